// DependencyParser_23021024707460
// MI455X (gfx1250) — compile-verified
//
#include <hip/hip_runtime.h>
#include <math.h>

// ---------------------------------------------------------------------------
// DependencyParser on gfx1250 (MI455X).
// All GEMMs use V_WMMA_F32_16X16X4_F32 (f32 in/out -> matches the f32
// reference; workload is latency-bound on the LSTM recurrence, so fp16
// WMMA rate would buy nothing).
// ---------------------------------------------------------------------------

typedef float v2f __attribute__((ext_vector_type(2)));
typedef float v8f __attribute__((ext_vector_type(8)));

#define WMMA_F32(a, b, c) \
    __builtin_amdgcn_wmma_f32_16x16x4_f32(false, (a), false, (b), (short)0, (c), false, false)

__device__ __forceinline__ float sigmoidf_(float x) { return 1.0f / (1.0f + expf(-x)); }

// ---------------------------------------------------------------------------
// K1: embedding gather -> xT (L, B, 128) time-major, f32
// ---------------------------------------------------------------------------
__global__ void dp_embed_kernel(const int* __restrict__ wi, const int* __restrict__ pi,
                                const float* __restrict__ we, const float* __restrict__ te,
                                float* __restrict__ xT) {
    int tid = blockIdx.x * 256 + threadIdx.x;          // 2048*128 threads exactly
    int d = tid & 127;
    int r = tid >> 7;            // r = t*16 + b
    int t = r >> 4;
    int b = r & 15;
    float v;
    if (d < 100) v = we[wi[b * 128 + t] * 100 + d];
    else         v = te[pi[b * 128 + t] * 28 + (d - 100)];
    xT[r * 128 + d] = v;
}

// ---------------------------------------------------------------------------
// K2/K4/K6: C[m][n] = sum_k A[m][k] * W[n][koff+k]  (+ bias0[n] + bias1[n])
// One wave per 16x16 tile. grid = (ceil(N/16), M/16), block = 32.
// Out-of-range n lanes (only the padded 100->112 case) are CLAMPED to a valid
// W row instead of zeroed: their D columns are never stored, and clamping
// keeps the inner loop free of divergent/masked loads (uniform b64 loads).
// WMMA f32 16x16x4 lane mapping:
//   A: lane&15 = M row, (lane>>4)*2 = K base -> v2f {A[m][k], A[m][k+1]}
//   B: lane&15 = N col, (lane>>4)*2 = K base -> v2f {B[k][n], B[k+1][n]}
//   D: VGPR v -> row v + 8*(lane>>4), col = lane&15
// ---------------------------------------------------------------------------
__global__ void dp_wmma_gemm_wt(const float* __restrict__ A, int lda,
                                const float* __restrict__ W, int ldw, int koff,
                                const float* __restrict__ bias0, const float* __restrict__ bias1,
                                float* __restrict__ C, int ldc, int Nvalid, int K) {
    const int lane = threadIdx.x;
    const int m0 = blockIdx.y * 16, n0 = blockIdx.x * 16;
    const int mn = lane & 15;
    const int half = lane >> 4;
    const int kb = half * 2;

    v8f acc = {0.f, 0.f, 0.f, 0.f, 0.f, 0.f, 0.f, 0.f};
    const float* Arow = A + (m0 + mn) * lda + kb;
    int nclamp = n0 + mn;
    if (nclamp > Nvalid - 1) nclamp = Nvalid - 1;     // uniform loads, cols dropped at store
    const float* Wrow = W + nclamp * ldw + koff + kb;

    for (int k0 = 0; k0 < K; k0 += 4) {
        v2f a = *(const v2f*)(Arow + k0);
        v2f b = *(const v2f*)(Wrow + k0);
        acc = WMMA_F32(a, b, acc);
    }

    const int col = n0 + mn;
    if (col < Nvalid) {
        float bs = 0.f;
        if (bias0) bs += bias0[col];
        if (bias1) bs += bias1[col];
#pragma unroll
        for (int v = 0; v < 8; ++v) {
            int row = m0 + v + 8 * half;
            C[row * ldc + col] = acc[v] + bs;
        }
    }
}

// ---------------------------------------------------------------------------
// K3/K5: LSTM recurrence, one block per direction (blockIdx.x: 0=fwd, 1=bwd).
// 256 threads = 8 waves. Per timestep: g(16x512) = h(16x128) @ whh^T + xp[t],
// done as 8 waves x 4 tiles x 32 WMMA k-chunks; then fused gate math in LDS.
// h,c state lives in LDS across the 128 steps. Output hout is (L, B, 256)
// with fwd in cols [0,128) and bwd in cols [128,256).
// ---------------------------------------------------------------------------
__global__ void dp_lstm_rec_kernel(const float* __restrict__ xp_f, const float* __restrict__ xp_b,
                                   const float* __restrict__ whh_f, const float* __restrict__ whh_b,
                                   float* __restrict__ hout) {
    __shared__ float h_s[16][132];   // +4 pad (even): bank-spread, keeps b64 alignment
    __shared__ float c_s[16][132];
    __shared__ float g_s[16][516];

    const int dir = blockIdx.x;
    const float* xp  = dir ? xp_b  : xp_f;
    const float* whh = dir ? whh_b : whh_f;
    const int tid = threadIdx.x;

    for (int i = tid; i < 16 * 128; i += 256) {
        h_s[i >> 7][i & 127] = 0.f;
        c_s[i >> 7][i & 127] = 0.f;
    }
    __syncthreads();

    const int lane = tid & 31;
    const int wave = tid >> 5;
    const int mn = lane & 15;
    const int half = lane >> 4;
    const int kb = half * 2;

    for (int step = 0; step < 128; ++step) {
        const int t = dir ? (127 - step) : step;

        // --- gate GEMM: g = h_s @ whh^T + xp[t]  (16 x 512, K=128) ---
        for (int tile = 0; tile < 4; ++tile) {
            const int n0 = wave * 64 + tile * 16;
            v8f acc = {0.f, 0.f, 0.f, 0.f, 0.f, 0.f, 0.f, 0.f};
            const float* Wrow = whh + (n0 + mn) * 128 + kb;
            const float* Hrow = &h_s[mn][kb];
#pragma unroll 4
            for (int k0 = 0; k0 < 128; k0 += 4) {
                v2f a = *(const v2f*)(Hrow + k0);
                v2f b = *(const v2f*)(Wrow + k0);
                acc = WMMA_F32(a, b, acc);
            }
            const int col = n0 + mn;
#pragma unroll
            for (int v = 0; v < 8; ++v) {
                int row = v + 8 * half;
                g_s[row][col] = acc[v] + xp[(t * 16 + row) * 512 + col];
            }
        }
        __syncthreads();

        // --- gate nonlinearity + state update (i,f,g,o split) ---
        for (int i = tid; i < 16 * 128; i += 256) {
            int b  = i >> 7;
            int hh = i & 127;
            float gi = g_s[b][hh];
            float gf = g_s[b][128 + hh];
            float gg = g_s[b][256 + hh];
            float go = g_s[b][384 + hh];
            float c = sigmoidf_(gf) * c_s[b][hh] + sigmoidf_(gi) * tanhf(gg);
            float h = sigmoidf_(go) * tanhf(c);
            c_s[b][hh] = c;
            h_s[b][hh] = h;
            hout[(t * 16 + b) * 256 + dir * 128 + hh] = h;
        }
        __syncthreads();
    }
}

// ---------------------------------------------------------------------------
// K7: scores[(i*128+j)*16 + b] = fc2_b + sum_k fc2_w[k]*tanh(a[i,b,k]+bp[j,b,k])
// ---------------------------------------------------------------------------
__global__ void dp_score_kernel(const float* __restrict__ aa, const float* __restrict__ bb,
                                const float* __restrict__ w2, const float* __restrict__ b2,
                                float* __restrict__ out) {
    int o = blockIdx.x * 256 + threadIdx.x;  // exactly 128*128*16 threads
    int b = o & 15;
    int j = (o >> 4) & 127;
    int i = o >> 11;
    const float* ai = aa + (i * 16 + b) * 100;
    const float* bj = bb + (j * 16 + b) * 100;
    float s = b2[0];
#pragma unroll 4
    for (int k = 0; k < 100; ++k) s += w2[k] * tanhf(ai[k] + bj[k]);
    out[o] = s;
}

// ---------------------------------------------------------------------------
// Launcher. Workspace layout (floats), total ~3.82M floats (~14.6 MB):
//   xT  (L,B,128)=262144 | xpf (L,B,512)=1048576 | xpb 1048576
//   h0  (L,B,256)=524288 | h1 524288 | abuf (L,B,100)=204800 | bbuf 204800
// xpf/xpb are reused between layers (stream-ordered).
// ---------------------------------------------------------------------------
extern "C" void kernel_launch(void* const* d_in, const int* in_sizes, int n_in,
                              void* d_out, int out_size, void* d_ws, size_t ws_size,
                              hipStream_t stream) {
    const int*   words   = (const int*)d_in[0];
    const int*   pos     = (const int*)d_in[1];
    // d_in[2] = max_length (always 128), d_in[3] = lengths (unused by reference)
    const float* we      = (const float*)d_in[4];
    const float* te      = (const float*)d_in[5];
    const float* w_ih_0f = (const float*)d_in[6];
    const float* w_hh_0f = (const float*)d_in[7];
    const float* b_ih_0f = (const float*)d_in[8];
    const float* b_hh_0f = (const float*)d_in[9];
    const float* w_ih_0b = (const float*)d_in[10];
    const float* w_hh_0b = (const float*)d_in[11];
    const float* b_ih_0b = (const float*)d_in[12];
    const float* b_hh_0b = (const float*)d_in[13];
    const float* w_ih_1f = (const float*)d_in[14];
    const float* w_hh_1f = (const float*)d_in[15];
    const float* b_ih_1f = (const float*)d_in[16];
    const float* b_hh_1f = (const float*)d_in[17];
    const float* w_ih_1b = (const float*)d_in[18];
    const float* w_hh_1b = (const float*)d_in[19];
    const float* b_ih_1b = (const float*)d_in[20];
    const float* b_hh_1b = (const float*)d_in[21];
    const float* fc1_w   = (const float*)d_in[22];
    const float* fc1_b   = (const float*)d_in[23];
    const float* fc2_w   = (const float*)d_in[24];
    const float* fc2_b   = (const float*)d_in[25];

    float* ws   = (float*)d_ws;
    float* xT   = ws;
    float* xpf  = xT  + 262144;
    float* xpb  = xpf + 1048576;
    float* h0   = xpb + 1048576;
    float* h1   = h0  + 524288;
    float* abuf = h1  + 524288;
    float* bbuf = abuf + 204800;

    // K1: embeddings
    dp_embed_kernel<<<1024, 256, 0, stream>>>(words, pos, we, te, xT);

    // K2: layer-0 input projections (M=2048, N=512, K=128)
    dim3 g512(32, 128);
    dp_wmma_gemm_wt<<<g512, 32, 0, stream>>>(xT, 128, w_ih_0f, 128, 0, b_ih_0f, b_hh_0f, xpf, 512, 512, 128);
    dp_wmma_gemm_wt<<<g512, 32, 0, stream>>>(xT, 128, w_ih_0b, 128, 0, b_ih_0b, b_hh_0b, xpb, 512, 512, 128);

    // K3: layer-0 recurrence (fwd + bwd) -> h0 (L,B,256)
    dp_lstm_rec_kernel<<<2, 256, 0, stream>>>(xpf, xpb, w_hh_0f, w_hh_0b, h0);

    // K4: layer-1 input projections (K=256)
    dp_wmma_gemm_wt<<<g512, 32, 0, stream>>>(h0, 256, w_ih_1f, 256, 0, b_ih_1f, b_hh_1f, xpf, 512, 512, 256);
    dp_wmma_gemm_wt<<<g512, 32, 0, stream>>>(h0, 256, w_ih_1b, 256, 0, b_ih_1b, b_hh_1b, xpb, 512, 512, 256);

    // K5: layer-1 recurrence -> h1 (L,B,256)
    dp_lstm_rec_kernel<<<2, 256, 0, stream>>>(xpf, xpb, w_hh_1f, w_hh_1b, h1);

    // K6: a = h1 @ wa^T, bp = h1 @ wb^T + fc1_b   (N=100 padded to 112)
    dim3 g100(7, 128);
    dp_wmma_gemm_wt<<<g100, 32, 0, stream>>>(h1, 256, fc1_w, 512, 0,   nullptr, nullptr, abuf, 100, 100, 256);
    dp_wmma_gemm_wt<<<g100, 32, 0, stream>>>(h1, 256, fc1_w, 512, 256, fc1_b,  nullptr, bbuf, 100, 100, 256);

    // K7: biaffine tanh scores -> out (16384, 16, 1) f32
    dp_score_kernel<<<1024, 256, 0, stream>>>(abuf, bbuf, fc2_w, fc2_b, (float*)d_out);
}